// RevSpatialTransformer_79611513799329
// MI455X (gfx1250) — compile-verified
//
#include <hip/hip_runtime.h>
#include <hip/hip_bf16.h>

typedef __attribute__((ext_vector_type(2))) float v2f;
typedef __attribute__((ext_vector_type(8))) float v8f;

#define NPTS  (32 * 32 * 32)   // 32768 grid points / warped points
#define NTILE (NPTS / 16)      // 2048 column tiles of 16 points
#define RAD2  10.24f           // (32/10)^2 squared search radius

// One wave handles TWO 16-query tiles of one batch.
// K=4 trick: A row = (-2qx,-2qy,-2qz, 1), B col = (px,py,pz, pn), so one
// V_WMMA_F32_16X16X4_F32 directly yields c = pn - 2 q.p for a 16x16 tile.
// The per-row constant qn doesn't change the argmin over columns, so it is
// added only at the end for the radius test. Per-tile VALU = cmp+select only.
__global__ __launch_bounds__(128)
void RevST_nn_wmma_kernel(const float* __restrict__ flow, float* __restrict__ out) {
    const int lane = threadIdx.x & 31;
    const int wave = threadIdx.x >> 5;
    const int gw   = blockIdx.x * 4 + wave;    // 0..2047
    const int b    = gw >> 10;                 // batch 0..1
    const int qp   = gw & 1023;                // query tile pair 0..1023
    const int half = lane >> 4;                // 0: rows K0/K1, 1: rows K2/K3
    const int lcol = lane & 15;

    const float* fb = flow + b * 3 * NPTS;     // flow[b, c, g] = fb[c*NPTS + g]

    // ---- A matrices: queries = integer grid, scaled by -2, K3 = 1 -------
    // lanes 0-15: (K0,K1) = (-2x,-2y) of query M=lane; lanes 16-31: (-2z, 1)
    v2f A[2];
#pragma unroll
    for (int s = 0; s < 2; ++s) {
        const int g = (qp * 2 + s) * 16 + lcol;
        const float ax = -2.0f * (float)(g >> 10);
        const float ay = -2.0f * (float)((g >> 5) & 31);
        const float az = -2.0f * (float)(g & 31);
        A[s].x = half ? az : ax;
        A[s].y = half ? 1.0f : ay;
    }

    float best[2][8];
    int   bidx[2][8];
#pragma unroll
    for (int s = 0; s < 2; ++s)
#pragma unroll
        for (int v = 0; v < 8; ++v) { best[s][v] = 3.4e38f; bidx[s][v] = 0; }

    // ---- scan all 2048 point tiles --------------------------------------
    for (int t = 0; t < NTILE; ++t) {
        const int j = t * 16 + lcol;           // this lane's point (column)
        // warped point = grid + flow  (3 coalesced 16-lane loads, L2-resident)
        const float px = (float)(j >> 10)       + fb[j];
        const float py = (float)((j >> 5) & 31) + fb[NPTS + j];
        const float pz = (float)(j & 31)        + fb[2 * NPTS + j];

        // prefetch next tile's rows (global_prefetch_b8; speculative, droppable)
        __builtin_prefetch(fb + j + 16);
        __builtin_prefetch(fb + NPTS + j + 16);
        __builtin_prefetch(fb + 2 * NPTS + j + 16);

        const float pn = fmaf(px, px, fmaf(py, py, pz * pz));

        // B 4x16: lanes 0-15 rows (K0,K1)=(px,py); lanes 16-31 rows (pz,pn)
        v2f B;
        B.x = half ? pz : px;
        B.y = half ? pn : py;

#pragma unroll
        for (int s = 0; s < 2; ++s) {
            v8f c = {};
            c = __builtin_amdgcn_wmma_f32_16x16x4_f32(
                    /*neg_a=*/false, A[s], /*neg_b=*/false, B,
                    /*c_mod=*/(short)0, c, /*reuse_a=*/false, /*reuse_b=*/false);
#pragma unroll
            for (int v = 0; v < 8; ++v) {
                const bool lt = c[v] < best[s][v];   // c[v] = pn - 2 q.p
                best[s][v] = lt ? c[v] : best[s][v];
                bidx[s][v] = lt ? j : bidx[s][v];
            }
        }
    }

    // ---- per query tile: argmin across 16 lanes of each half, then write
#pragma unroll
    for (int s = 0; s < 2; ++s) {
        const int qt = qp * 2 + s;
#pragma unroll
        for (int v = 0; v < 8; ++v) {
            float d = best[s][v];
            int   i = bidx[s][v];
#pragma unroll
            for (int m = 1; m < 16; m <<= 1) {
                const float od = __shfl_xor(d, m, 32);
                const int   oi = __shfl_xor(i, m, 32);
                if (od < d || (od == d && oi < i)) { d = od; i = oi; }
            }
            best[s][v] = d;
            bidx[s][v] = i;
        }

        // lane 0 writes rows M=0..7 (half 0), lane 16 writes M=8..15
        if (lcol == 0) {
#pragma unroll
            for (int v = 0; v < 8; ++v) {
                const int M = v + 8 * half;
                const int g = qt * 16 + M;
                const float qx = (float)(g >> 10);
                const float qy = (float)((g >> 5) & 31);
                const float qz = (float)(g & 31);
                const float qn = qx * qx + qy * qy + qz * qz;
                const int i = bidx[s][v];
                const bool valid = (best[s][v] + qn) <= RAD2;  // true d2 <= r^2
                const float ox = valid ? -fb[i]            : 0.0f;
                const float oy = valid ? -fb[NPTS + i]     : 0.0f;
                const float oz = valid ? -fb[2 * NPTS + i] : 0.0f;
                float* ob = out + b * 3 * NPTS;
                ob[0 * NPTS + g] = ox;
                ob[1 * NPTS + g] = oy;
                ob[2 * NPTS + g] = oz;
            }
        }
    }
}

extern "C" void kernel_launch(void* const* d_in, const int* in_sizes, int n_in,
                              void* d_out, int out_size, void* d_ws, size_t ws_size,
                              hipStream_t stream) {
    (void)in_sizes; (void)n_in; (void)d_ws; (void)ws_size; (void)out_size;
    const float* flow = (const float*)d_in[0];   // (2, 3, 32, 32, 32) f32
    float* out = (float*)d_out;                  // (2, 3, 32, 32, 32) f32

    // 2 batches * 1024 tile-pairs = 2048 waves; 4 waves per 128-thread block
    const int nWaves = 2 * (NTILE / 2);
    const int blocks = nWaves / 4;               // 512
    RevST_nn_wmma_kernel<<<blocks, 128, 0, stream>>>(flow, out);
}